// BktModel_29892972380908
// MI455X (gfx1250) — compile-verified
//
#include <hip/hip_runtime.h>
#include <math.h>

// ---------------------------------------------------------------------------
// BKT HMM forward for MI455X (gfx1250, wave32).
//   B=1024, T=500 (sequential scan), C=64, K=2000, S=O=2.
// One wave (32 lanes) owns a 16-batch tile; log_alpha lives in registers in
// WMMA-B fragment layout so that
//   * a2[b,s] = diag( cc[16x64] x la[64x16,s] )  -> V_WMMA_F32_16X16X4_F32
//   * a1/a2c  =       cc[16x64] x W[64x16]       -> V_WMMA_F32_16X16X4_F32
//   * state update la = fma(cc, a3-la, la) is register-local (A/B fragments
//     of 16x16x4 f32 place the same (b,c) element in the same lane/slot).
// Software-pipelined: the A-row gather for step t+1 is issued before the
// WMMA + logsumexp work of step t, hiding L2 gather latency behind compute.
// ---------------------------------------------------------------------------

typedef float v2f __attribute__((ext_vector_type(2)));
typedef float v8f __attribute__((ext_vector_type(8)));

#define BSZ 1024
#define TSZ 500
#define CSZ 64
// S = O = 2 hard-coded.

// Workspace layout in d_ws (floats):
//   [0 .. 64*16)         W[c][n]: n=0..3 -> log_obs[c][s*2+o],
//                                 n=4..7 -> log_t[c][s*2+t'], n=8..15 -> 0
//   [64*16 .. 64*16+128) linit[c][s] = log_softmax(init_logits[c], axis=s)

__device__ __forceinline__ float lse2(float x, float y) {
    float m = fmaxf(x, y);
    return m + __logf(__expf(x - m) + __expf(y - m));
}

__global__ __launch_bounds__(64) void bkt_prep(const float* __restrict__ trans,
                                               const float* __restrict__ obs,
                                               const float* __restrict__ init,
                                               float* __restrict__ ws) {
    int c = threadIdx.x;
    if (c >= CSZ) return;
    float* W = ws;
    float* linit = ws + CSZ * 16;

    // obs_logits [C,S,O] -> log_softmax over O (axis 2)
    for (int s = 0; s < 2; ++s) {
        float x0 = obs[c * 4 + s * 2 + 0];
        float x1 = obs[c * 4 + s * 2 + 1];
        float z = lse2(x0, x1);
        W[c * 16 + s * 2 + 0] = x0 - z;
        W[c * 16 + s * 2 + 1] = x1 - z;
    }
    // trans_logits [C,S,S] -> log_softmax over axis 1 (first state index)
    for (int tp = 0; tp < 2; ++tp) {
        float x0 = trans[c * 4 + 0 * 2 + tp];
        float x1 = trans[c * 4 + 1 * 2 + tp];
        float z = lse2(x0, x1);
        W[c * 16 + 4 + 0 * 2 + tp] = x0 - z;
        W[c * 16 + 4 + 1 * 2 + tp] = x1 - z;
    }
    for (int n = 8; n < 16; ++n) W[c * 16 + n] = 0.0f;
    // init_logits [C,S] -> log_softmax over S
    {
        float x0 = init[c * 2 + 0];
        float x1 = init[c * 2 + 1];
        float z = lse2(x0, x1);
        linit[c * 2 + 0] = x0 - z;
        linit[c * 2 + 1] = x1 - z;
    }
}

__global__ __launch_bounds__(32) void bkt_fwd(const int* __restrict__ corr,
                                              const int* __restrict__ kc,
                                              const float* __restrict__ Amat,
                                              const float* __restrict__ ws,
                                              float* __restrict__ out) {
    const int lane = threadIdx.x;      // 0..31
    const int bl   = lane & 15;        // local batch row (WMMA M / N index)
    const int half = lane >> 4;        // K-offset selector within a K=4 chunk
    const int bg   = blockIdx.x * 16 + bl;  // global batch id

    __shared__ float sAccW[16 * 16];
    __shared__ float sAccS[2][16 * 16];
    __shared__ float sA3[16][2];

    const float* W     = ws;
    const float* linit = ws + CSZ * 16;

    // --- Preload W into B-fragments (invariant over T). Chunk k covers
    // c = 4k..4k+3; lane holds B[K = j + 2*half][N = bl].
    float Wf[16][2];
#pragma unroll
    for (int k = 0; k < 16; ++k) {
        int c = 4 * k + 2 * half;
        Wf[k][0] = W[(c + 0) * 16 + bl];
        Wf[k][1] = W[(c + 1) * 16 + bl];
    }
    // --- Init log_alpha in B-fragment layout: la[s][k][j] = la[b=bl][c][s]
    float la[2][16][2];
#pragma unroll
    for (int k = 0; k < 16; ++k) {
        int c = 4 * k + 2 * half;
#pragma unroll
        for (int j = 0; j < 2; ++j) {
            la[0][k][j] = linit[(c + j) * 2 + 0];
            la[1][k][j] = linit[(c + j) * 2 + 1];
        }
    }

    // Gather of one A row into A-fragment layout:
    // lane holds A[M = bl][K = j + 2*half] for chunk k (c = 4k + j + 2*half).
    auto gather = [&](float (&dst)[16][2], int kcvv) {
#pragma unroll
        for (int k = 0; k < 16; ++k) {
            const float2 v = *(const float2*)(Amat + kcvv * CSZ + 4 * k + 2 * half);
            dst[k][0] = v.x;
            dst[k][1] = v.y;
        }
    };

    // One HMM step: emit log_py[t], update la in-place. cc = fragments for t.
    auto compute = [&](int t, const float (&cc)[16][2], int y) {
        v8f accW = {};  // [b][n]: n=0..3 -> a1[s][o], n=4..7 -> a2c[s][t']
        v8f acc0 = {};  // [b][b']: diag = a2[b][s=0]
        v8f acc1 = {};  // [b][b']: diag = a2[b][s=1]
#pragma unroll
        for (int k = 0; k < 16; ++k) {
            v2f a  = {cc[k][0], cc[k][1]};
            v2f bw = {Wf[k][0], Wf[k][1]};
            v2f b0 = {la[0][k][0], la[0][k][1]};
            v2f b1 = {la[1][k][0], la[1][k][1]};
            accW = __builtin_amdgcn_wmma_f32_16x16x4_f32(false, a, false, bw, (short)0, accW, false, false);
            acc0 = __builtin_amdgcn_wmma_f32_16x16x4_f32(false, a, false, b0, (short)0, acc0, false, false);
            acc1 = __builtin_amdgcn_wmma_f32_16x16x4_f32(false, a, false, b1, (short)0, acc1, false, false);
        }

        // Dump C-fragments (lane holds rows m = r + 8*half, col = bl).
#pragma unroll
        for (int r = 0; r < 8; ++r) {
            sAccW[(r + 8 * half) * 16 + bl]    = accW[r];
            sAccS[0][(r + 8 * half) * 16 + bl] = acc0[r];
            sAccS[1][(r + 8 * half) * 16 + bl] = acc1[r];
        }
        __syncthreads();

        if (lane < 16) {
            const int b = lane;
            const float a1_00 = sAccW[b * 16 + 0];  // a1[s=0][o=0]
            const float a1_01 = sAccW[b * 16 + 1];
            const float a1_10 = sAccW[b * 16 + 2];
            const float a1_11 = sAccW[b * 16 + 3];
            const float c_00  = sAccW[b * 16 + 4];  // a2c[s=0][t'=0]
            const float c_01  = sAccW[b * 16 + 5];
            const float c_10  = sAccW[b * 16 + 6];
            const float c_11  = sAccW[b * 16 + 7];
            const float a2_0  = sAccS[0][b * 16 + b];
            const float a2_1  = sAccS[1][b * 16 + b];

            // log_py[o] = logsoftmax_o( LSE_s(a1[s][o] + a2[s]) )
            float p0 = lse2(a1_00 + a2_0, a1_10 + a2_1);
            float p1 = lse2(a1_01 + a2_0, a1_11 + a2_1);
            float z  = lse2(p0, p1);
            float2 o2;
            o2.x = p0 - z;
            o2.y = p1 - z;
            *(float2*)(out + (size_t)(bg * TSZ + t) * 2) = o2;

            // lp[t'] = a1[t'][y];   a3[s] = LSE_t'(lp[t'] + a2[t'] + a2c[s][t'])
            const float lp0 = y ? a1_01 : a1_00;
            const float lp1 = y ? a1_11 : a1_10;
            sA3[b][0] = lse2(lp0 + a2_0 + c_00, lp1 + a2_1 + c_01);
            sA3[b][1] = lse2(lp0 + a2_0 + c_10, lp1 + a2_1 + c_11);
        }
        __syncthreads();

        // Register-local soft update: la += cc * (a3 - la).
        const float a3s0 = sA3[bl][0];
        const float a3s1 = sA3[bl][1];
#pragma unroll
        for (int k = 0; k < 16; ++k) {
#pragma unroll
            for (int j = 0; j < 2; ++j) {
                la[0][k][j] = fmaf(cc[k][j], a3s0 - la[0][k][j], la[0][k][j]);
                la[1][k][j] = fmaf(cc[k][j], a3s1 - la[1][k][j], la[1][k][j]);
            }
        }
    };

    // ---- Software pipeline (manual unroll-by-2; TSZ is even) ----
    // Scalars: (y_c) for current step, (kcv_n, y_n) for the following step.
    float ccA[16][2], ccB[16][2];
    int kcv_c = kc[bg * TSZ + 0];
    int y_c   = corr[bg * TSZ + 0];
    int kcv_n = kc[bg * TSZ + 1];
    int y_n   = corr[bg * TSZ + 1];
    gather(ccA, kcv_c);

    for (int t = 0; t < TSZ; t += 2) {
        // Even step t: prefetch gather for t+1, indices for t+2, compute t.
        gather(ccB, kcv_n);
        const int i2 = (t + 2 < TSZ) ? t + 2 : TSZ - 1;
        const int kcv2 = kc[bg * TSZ + i2];
        const int y2   = corr[bg * TSZ + i2];
        compute(t, ccA, y_c);

        // Odd step t+1: prefetch gather for t+2, indices for t+3, compute t+1.
        gather(ccA, kcv2);                       // clamped (wasted) on last iter
        const int i3 = (t + 3 < TSZ) ? t + 3 : TSZ - 1;
        const int kcv3 = kc[bg * TSZ + i3];
        const int y3   = corr[bg * TSZ + i3];
        compute(t + 1, ccB, y_n);

        y_c = y2;
        kcv_n = kcv3;
        y_n = y3;
    }
}

extern "C" void kernel_launch(void* const* d_in, const int* in_sizes, int n_in,
                              void* d_out, int out_size, void* d_ws, size_t ws_size,
                              hipStream_t stream) {
    const int*   corr  = (const int*)d_in[0];    // [B,T] int32
    const int*   kc    = (const int*)d_in[1];    // [B,T] int32
    const float* Amat  = (const float*)d_in[2];  // [K,C] f32
    const float* trans = (const float*)d_in[3];  // [C,S,S] f32
    const float* obs   = (const float*)d_in[4];  // [C,S,O] f32
    const float* init  = (const float*)d_in[5];  // [C,S] f32
    float*       out   = (float*)d_out;          // [B,T,O] f32
    float*       ws    = (float*)d_ws;           // >= (64*16 + 128) floats

    (void)in_sizes; (void)n_in; (void)out_size; (void)ws_size;

    bkt_prep<<<1, 64, 0, stream>>>(trans, obs, init, ws);
    bkt_fwd<<<BSZ / 16, 32, 0, stream>>>(corr, kc, Amat, ws, out);
}